// FastFK_56942676410822
// MI455X (gfx1250) — compile-verified
//
#include <hip/hip_runtime.h>

#define NJ 24
#define NV 6890
#define NB 10
#define BATCH 8192

typedef __attribute__((ext_vector_type(2))) float v2f;
typedef __attribute__((ext_vector_type(8))) float v8f;

static constexpr int kParents[NJ] = {-1, 0, 0, 0, 1, 2, 3, 4, 5, 6, 7, 8,
                                     9, 9, 9, 12, 13, 14, 16, 17, 18, 19, 20, 21};

// ---------------------------------------------------------------------------
// Kernel 1: batch-independent collapse of the vertex dimension.
//   JC[l][n] (12 x 80, row-major, padded) with n = j*3+k (n<72), l<10: beta dirs,
//   l==10: template row (multiplied by constant 1), l==11 & n>=72: zero padding.
//   JC[l][n] = sum_m Jreg[j,m] * shapedirs[m,k,l]   (or v_template[m,k] for l==10)
// ---------------------------------------------------------------------------
__global__ void fk_precompute(const float* __restrict__ v_template,
                              const float* __restrict__ shapedirs,
                              const float* __restrict__ Jreg,
                              float* __restrict__ JC) {
  int t = blockIdx.x * blockDim.x + threadIdx.x;
  if (t >= 12 * 80) return;
  int l = t / 80, n = t % 80;
  float acc = 0.0f;
  if (l < 11 && n < 72) {
    int j = n / 3, k = n % 3;
    const float* __restrict__ wrow = Jreg + j * NV;
    if (l < 10) {
      for (int m = 0; m < NV; ++m)
        acc += wrow[m] * shapedirs[(m * 3 + k) * NB + l];
    } else {
      for (int m = 0; m < NV; ++m)
        acc += wrow[m] * v_template[m * 3 + k];
    }
  }
  JC[t] = acc;  // padding rows/cols get exact zeros
}

// ---------------------------------------------------------------------------
// Kernel 2: WMMA joint regression + per-sample forward kinematics.
//   128 threads = 4 waves per block; block handles 128 batch samples.
// ---------------------------------------------------------------------------
__global__ __launch_bounds__(128) void fk_main(const float* __restrict__ pose6d,
                                               const float* __restrict__ betas,
                                               const float* __restrict__ JC,
                                               float* __restrict__ out) {
  __shared__ float sJC[12][80];   // K x N operand (B matrix source)
  __shared__ float sJ[128][72];   // per-sample rest joints J[b][j*3+k]

  const int tid  = threadIdx.x;
  const int b0   = blockIdx.x * 128;
  const int lane = tid & 31;
  const int wave = tid >> 5;

  // Stage JC into LDS.
  for (int i = tid; i < 12 * 80; i += 128) sJC[i / 80][i % 80] = JC[i];

  // Prefetch this thread's pose row while we do the WMMA phase.
  __builtin_prefetch(pose6d + (size_t)(b0 + tid) * (NJ * 6), 0, 0);

  __syncthreads();

  // ---- Phase 2: J = betasExt[128x12] @ JC[12x80] via V_WMMA_F32_16X16X4_F32.
  // A 16x4 f32 layout: lanes 0-15 hold K0(K2 for hi lanes) in v0, K1(K3) in v1.
  const int mrow  = lane & 15;
  const int khalf = lane >> 4;

  #pragma unroll
  for (int mi = 0; mi < 2; ++mi) {
    const int mt = wave * 2 + mi;            // M-tile: local rows mt*16..mt*16+15
    const int brow = b0 + mt * 16 + mrow;    // global batch row this lane owns
    const float* __restrict__ beta_row = betas + (size_t)brow * NB;

    v2f afrag[3];
    #pragma unroll
    for (int ks = 0; ks < 3; ++ks) {
      const int k0 = ks * 4 + khalf * 2;
      const int k1 = k0 + 1;
      afrag[ks].x = (k0 < NB) ? beta_row[k0] : (k0 == NB ? 1.0f : 0.0f);
      afrag[ks].y = (k1 < NB) ? beta_row[k1] : (k1 == NB ? 1.0f : 0.0f);
    }

    #pragma unroll
    for (int nt = 0; nt < 5; ++nt) {
      v8f acc = {0.f, 0.f, 0.f, 0.f, 0.f, 0.f, 0.f, 0.f};
      #pragma unroll
      for (int ks = 0; ks < 3; ++ks) {
        const int k0 = ks * 4 + khalf * 2;
        v2f bfrag;
        bfrag.x = sJC[k0][nt * 16 + mrow];
        bfrag.y = sJC[k0 + 1][nt * 16 + mrow];
        acc = __builtin_amdgcn_wmma_f32_16x16x4_f32(
            /*neg_a=*/false, afrag[ks], /*neg_b=*/false, bfrag,
            /*c_mod=*/(short)0, acc, /*reuse_a=*/false, /*reuse_b=*/false);
      }
      // D 16x16 f32 layout: lane n -> column n; VGPR r -> row r (+8 for hi lanes).
      const int col = nt * 16 + mrow;
      if (col < 72) {
        #pragma unroll
        for (int r = 0; r < 8; ++r)
          sJ[mt * 16 + r + 8 * khalf][col] = acc[r];
      }
    }
  }
  __syncthreads();

  // ---- Phase 3: per-sample FK chain (fully unrolled, compile-time parents).
  const int b = b0 + tid;
  const float* __restrict__ pose = pose6d + (size_t)b * (NJ * 6);
  float* __restrict__ o = out + (size_t)b * (NJ * 3);

  float Rw[NJ][9];
  float tw[NJ][3];

  #pragma unroll
  for (int i = 0; i < NJ; ++i) {
    // rot6d -> rotation matrix (rows b1, b2, b3)
    const float a1x = pose[i * 6 + 0], a1y = pose[i * 6 + 1], a1z = pose[i * 6 + 2];
    const float a2x = pose[i * 6 + 3], a2y = pose[i * 6 + 4], a2z = pose[i * 6 + 5];
    const float inv1 = rsqrtf(a1x * a1x + a1y * a1y + a1z * a1z);
    const float b1x = a1x * inv1, b1y = a1y * inv1, b1z = a1z * inv1;
    const float d = b1x * a2x + b1y * a2y + b1z * a2z;
    const float c2x = a2x - d * b1x, c2y = a2y - d * b1y, c2z = a2z - d * b1z;
    const float inv2 = rsqrtf(c2x * c2x + c2y * c2y + c2z * c2z);
    const float b2x = c2x * inv2, b2y = c2y * inv2, b2z = c2z * inv2;
    const float b3x = b1y * b2z - b1z * b2y;
    const float b3y = b1z * b2x - b1x * b2z;
    const float b3z = b1x * b2y - b1y * b2x;
    const float R[9] = {b1x, b1y, b1z, b2x, b2y, b2z, b3x, b3y, b3z};

    if (i == 0) {
      #pragma unroll
      for (int q = 0; q < 9; ++q) Rw[0][q] = R[q];
      #pragma unroll
      for (int k = 0; k < 3; ++k) tw[0][k] = sJ[tid][k];
    } else {
      const int p = kParents[i];
      float rel[3];
      #pragma unroll
      for (int k = 0; k < 3; ++k)
        rel[k] = sJ[tid][i * 3 + k] - sJ[tid][p * 3 + k];
      #pragma unroll
      for (int r = 0; r < 3; ++r) {
        #pragma unroll
        for (int c = 0; c < 3; ++c)
          Rw[i][r * 3 + c] = Rw[p][r * 3 + 0] * R[0 + c] +
                             Rw[p][r * 3 + 1] * R[3 + c] +
                             Rw[p][r * 3 + 2] * R[6 + c];
        tw[i][r] = Rw[p][r * 3 + 0] * rel[0] +
                   Rw[p][r * 3 + 1] * rel[1] +
                   Rw[p][r * 3 + 2] * rel[2] + tw[p][r];
      }
    }
    o[i * 3 + 0] = tw[i][0];
    o[i * 3 + 1] = tw[i][1];
    o[i * 3 + 2] = tw[i][2];
  }
}

// ---------------------------------------------------------------------------
extern "C" void kernel_launch(void* const* d_in, const int* in_sizes, int n_in,
                              void* d_out, int out_size, void* d_ws, size_t ws_size,
                              hipStream_t stream) {
  const float* pose6d     = (const float*)d_in[0];
  const float* betas      = (const float*)d_in[1];
  // d_in[2] = transl: unused by the reference output
  const float* v_template = (const float*)d_in[3];
  const float* shapedirs  = (const float*)d_in[4];
  const float* Jreg       = (const float*)d_in[5];
  float* JC  = (float*)d_ws;            // 12*80 floats = 3840 B of scratch
  float* out = (float*)d_out;           // 8192*24*3 floats

  fk_precompute<<<4, 256, 0, stream>>>(v_template, shapedirs, Jreg, JC);
  fk_main<<<BATCH / 128, 128, 0, stream>>>(pose6d, betas, JC, out);
}